// LocalAttention2d_44719199486641
// MI455X (gfx1250) — compile-verified
//
#include <hip/hip_runtime.h>
#include <hip/hip_bf16.h>
#include <math.h>

// Problem constants (from reference)
#define Bn    8
#define CIN   256
#define Hh    32
#define Ww    32
#define HW    1024
#define COUT  768        // KS*3*NH
#define KTAP  5
#define PADn  2
#define NHn   8
#define KSn   32
#define Gn    24         // 3*NH
#define NTAP  25         // KTAP*KTAP

typedef float v2f __attribute__((ext_vector_type(2)));
typedef float v8f __attribute__((ext_vector_type(8)));

// ---------------------------------------------------------------------------
// Kernel 1: 1x1 conv as GEMM with V_WMMA_F32_16X16X4_F32 (fp32 matrix pipe).
// attn[b, o, hw] = sum_c x[b, c, hw] * w[o, c] + bias[o]
// One wave per 16(M=o) x 64(N=hw) tile; A tile reused across 4 N-tiles.
//
// f32 WMMA operand layouts (wave32):
//   A (16x4, MxK):  lane l: row M = l&15, holds K = k0 + 2*(l>>4) + {0,1}
//   B (4x16, KxN):  lane l: col N = l&15, holds K = k0 + 2*(l>>4) + {0,1}
//   C/D (16x16):    vgpr i, lane l: M = i + 8*(l>>4), N = l&15
// ---------------------------------------------------------------------------
__global__ __launch_bounds__(32)
void conv_gemm_wmma(const float* __restrict__ x,
                    const float* __restrict__ w,
                    const float* __restrict__ bias,
                    float* __restrict__ attn)
{
    const int lane = threadIdx.x & 31;
    const int row  = lane & 15;
    const int half = lane >> 4;

    const int oT = blockIdx.x;   // 0..47  : Cout/16
    const int nG = blockIdx.y;   // 0..15  : groups of 4 N-tiles (64 cols)
    const int b  = blockIdx.z;   // 0..7

    const float* __restrict__ Wr = w + (size_t)(oT * 16 + row) * CIN + 2 * half;
    const float* __restrict__ Xb = x + (size_t)b * CIN * HW;

    const int n0 = (nG * 4 + 0) * 16 + row;
    const int n1 = n0 + 16;
    const int n2 = n0 + 32;
    const int n3 = n0 + 48;

    v8f acc0 = {}, acc1 = {}, acc2 = {}, acc3 = {};

    #pragma unroll 4
    for (int k0 = 0; k0 < CIN; k0 += 4) {
        v2f a;
        a.x = Wr[k0];
        a.y = Wr[k0 + 1];

        const float* __restrict__ Xk = Xb + (size_t)(k0 + 2 * half) * HW;

        v2f b0, b1, b2, b3;
        b0.x = Xk[n0]; b0.y = Xk[n0 + HW];
        b1.x = Xk[n1]; b1.y = Xk[n1 + HW];
        b2.x = Xk[n2]; b2.y = Xk[n2 + HW];
        b3.x = Xk[n3]; b3.y = Xk[n3 + HW];

        acc0 = __builtin_amdgcn_wmma_f32_16x16x4_f32(false, a, false, b0, (short)0, acc0, false, false);
        acc1 = __builtin_amdgcn_wmma_f32_16x16x4_f32(false, a, false, b1, (short)0, acc1, false, false);
        acc2 = __builtin_amdgcn_wmma_f32_16x16x4_f32(false, a, false, b2, (short)0, acc2, false, false);
        acc3 = __builtin_amdgcn_wmma_f32_16x16x4_f32(false, a, false, b3, (short)0, acc3, false, false);
    }

    #pragma unroll
    for (int i = 0; i < 8; ++i) {
        const int o  = oT * 16 + i + 8 * half;
        const float bv = bias[o];
        float* __restrict__ dst = attn + ((size_t)(b * COUT + o)) * HW;
        dst[n0] = acc0[i] + bv;
        dst[n1] = acc1[i] + bv;
        dst[n2] = acc2[i] + bv;
        dst[n3] = acc3[i] + bv;
    }
}

// ---------------------------------------------------------------------------
// Kernel 2: GroupNorm stats per (b, g, kh, kw).
// The 25 tiles are zero-padded shifts of attn, so mean/var over (KS,OH,OW)
// reduce to clipped-window sums of the per-group channel sums / sq-sums.
// One block per (b,g): O(HW) work instead of 25x re-reads.
// ---------------------------------------------------------------------------
__global__ __launch_bounds__(256)
void gn_stats(const float* __restrict__ attn,
              float* __restrict__ meanArr,
              float* __restrict__ rstdArr)
{
    __shared__ float P[HW];           // per-pixel channel-group sum
    __shared__ float Q[HW];           // per-pixel channel-group sq-sum
    __shared__ float rowP[KTAP][Hh];  // per-kw clipped row partials
    __shared__ float rowQ[KTAP][Hh];

    const int g = blockIdx.x;   // 0..23
    const int b = blockIdx.y;   // 0..7
    const int tid = threadIdx.x;

    const float* __restrict__ src = attn + ((size_t)(b * COUT + g * KSn)) * HW;

    for (int pix = tid; pix < HW; pix += 256) {
        float s = 0.f, s2 = 0.f;
        #pragma unroll
        for (int c = 0; c < KSn; ++c) {
            float v = src[(size_t)c * HW + pix];
            s  += v;
            s2 += v * v;
        }
        P[pix] = s;
        Q[pix] = s2;
    }
    __syncthreads();

    if (tid < KTAP * Hh) {   // 160 tasks: (kw, y)
        const int kw = tid >> 5;
        const int y  = tid & 31;
        const int c0 = (kw >= PADn) ? (kw - PADn) : 0;
        const int c1 = (kw + 29 < 31) ? (kw + 29) : 31;
        float s = 0.f, s2 = 0.f;
        for (int xc = c0; xc <= c1; ++xc) {
            s  += P[y * Ww + xc];
            s2 += Q[y * Ww + xc];
        }
        rowP[kw][y] = s;
        rowQ[kw][y] = s2;
    }
    __syncthreads();

    if (tid < NTAP) {
        const int kh = tid / KTAP;
        const int kw = tid % KTAP;
        const int r0 = (kh >= PADn) ? (kh - PADn) : 0;
        const int r1 = (kh + 29 < 31) ? (kh + 29) : 31;
        float s = 0.f, s2 = 0.f;
        for (int y = r0; y <= r1; ++y) {
            s  += rowP[kw][y];
            s2 += rowQ[kw][y];
        }
        const float inv  = 1.0f / 32768.0f;   // KS*OH*OW (zeros included)
        const float mean = s * inv;
        const float var  = s2 * inv - mean * mean;
        const size_t idx = (size_t)(b * Gn + g) * NTAP + tid;
        meanArr[idx] = mean;
        rstdArr[idx] = rsqrtf(var + 1e-5f);
    }
}

// ---------------------------------------------------------------------------
// Kernel 3: fused GN-apply + local attention.
// One block per (b, head, oh). Stage 96 channels x 5 rows x 32 cols of attn
// into LDS (60 KB); each wave32 handles 4 ow positions, lane = channel.
// Softmax over the 25 taps fully via wave32 shuffles (no block barriers in
// the compute loop).
// ---------------------------------------------------------------------------
__device__ __forceinline__ float wave_sum(float v) {
    #pragma unroll
    for (int o = 16; o > 0; o >>= 1) v += __shfl_xor(v, o, 32);
    return v;
}
__device__ __forceinline__ float wave_max(float v) {
    #pragma unroll
    for (int o = 16; o > 0; o >>= 1) v = fmaxf(v, __shfl_xor(v, o, 32));
    return v;
}

__global__ __launch_bounds__(256)
void local_attn(const float* __restrict__ attn,
                const float* __restrict__ meanArr,
                const float* __restrict__ rstdArr,
                const float* __restrict__ gn_w,
                const float* __restrict__ gn_b,
                float* __restrict__ out)
{
    __shared__ float tile[96 * KTAP * Ww];   // [lc][kh][col], 61440 B
    __shared__ float meanS[3 * NTAP];        // [rk][p], rk: 0=key,1=q,2=v
    __shared__ float rstdS[3 * NTAP];

    const int tid = threadIdx.x;
    const int oh  = blockIdx.x;   // 0..31
    const int n   = blockIdx.y;   // 0..7 head
    const int b   = blockIdx.z;   // 0..7

    // Stage the 5-row slab of this head's 96 channels (OOB rows -> 0).
    const float* __restrict__ src = attn + ((size_t)(b * COUT + 96 * n)) * HW;
    for (int i = tid; i < 96 * KTAP * Ww; i += 256) {
        const int lc  = i / (KTAP * Ww);
        const int rem = i - lc * (KTAP * Ww);
        const int kh  = rem >> 5;
        const int col = rem & 31;
        const int r   = oh + kh - PADn;
        tile[i] = (r >= 0 && r < Hh) ? src[(size_t)lc * HW + r * Ww + col] : 0.0f;
    }
    if (tid < 3 * NTAP) {
        const int rk = tid / NTAP;         // key / q / v sub-group
        const int p  = tid % NTAP;
        const int g  = 3 * n + rk;
        const size_t idx = (size_t)(b * Gn + g) * NTAP + p;
        meanS[tid] = meanArr[idx];
        rstdS[tid] = rstdArr[idx];
    }
    __syncthreads();

    const int wave = tid >> 5;
    const int lane = tid & 31;

    const float gwK = gn_w[96 * n + lane],      gbK = gn_b[96 * n + lane];
    const float gwQ = gn_w[96 * n + 32 + lane], gbQ = gn_b[96 * n + 32 + lane];
    const float gwV = gn_w[96 * n + 64 + lane], gbV = gn_b[96 * n + 64 + lane];

    const float* __restrict__ tK = tile + (size_t)lane * (KTAP * Ww);
    const float* __restrict__ tQ = tile + (size_t)(32 + lane) * (KTAP * Ww);
    const float* __restrict__ tV = tile + (size_t)(64 + lane) * (KTAP * Ww);

    #pragma unroll
    for (int i = 0; i < 4; ++i) {
        const int ow = wave * 4 + i;

        // q_s = mean over 25 taps of normalized q-channel (lane = channel s)
        float q = 0.f;
        #pragma unroll
        for (int p = 0; p < NTAP; ++p) {
            const int kh = p / KTAP, kw = p % KTAP;
            const int col = ow + kw - PADn;
            const float raw = (col >= 0 && col < Ww) ? tQ[kh * Ww + col] : 0.f;
            q += (raw - meanS[NTAP + p]) * rstdS[NTAP + p] * gwQ + gbQ;
        }
        q *= (1.0f / 25.0f);

        // sal_p = dot(q, key[:,p]) / sqrt(KS), result kept in lane p
        float mysal = -1e30f;
        #pragma unroll
        for (int p = 0; p < NTAP; ++p) {
            const int kh = p / KTAP, kw = p % KTAP;
            const int col = ow + kw - PADn;
            const float raw = (col >= 0 && col < Ww) ? tK[kh * Ww + col] : 0.f;
            const float keyv = (raw - meanS[p]) * rstdS[p] * gwK + gbK;
            const float s = wave_sum(q * keyv) * 0.17677669529663687f; // 1/sqrt(32)
            if (lane == p) mysal = s;
        }

        // softmax over lanes 0..24
        const float mx = wave_max(mysal);
        const float e  = (lane < NTAP) ? __expf(mysal - mx) : 0.f;
        const float denom = wave_sum(e);
        const float mask  = e / denom;

        // out_s = sum_p v[s,p] * mask[p]
        float acc = 0.f;
        #pragma unroll
        for (int p = 0; p < NTAP; ++p) {
            const int kh = p / KTAP, kw = p % KTAP;
            const int col = ow + kw - PADn;
            const float raw = (col >= 0 && col < Ww) ? tV[kh * Ww + col] : 0.f;
            const float vv  = (raw - meanS[2 * NTAP + p]) * rstdS[2 * NTAP + p] * gwV + gbV;
            acc += vv * __shfl(mask, p, 32);
        }

        out[((size_t)(b * (NHn * KSn) + n * KSn + lane)) * HW + oh * Ww + ow] = acc;
    }
}

// ---------------------------------------------------------------------------
extern "C" void kernel_launch(void* const* d_in, const int* in_sizes, int n_in,
                              void* d_out, int out_size, void* d_ws, size_t ws_size,
                              hipStream_t stream)
{
    const float* x      = (const float*)d_in[0];
    const float* conv_w = (const float*)d_in[1];
    const float* conv_b = (const float*)d_in[2];
    const float* gn_w   = (const float*)d_in[3];
    const float* gn_b   = (const float*)d_in[4];
    float* out = (float*)d_out;

    float* attn  = (float*)d_ws;                          // B*COUT*HW floats
    float* meanA = attn + (size_t)Bn * COUT * HW;         // B*G*25
    float* rstdA = meanA + (size_t)Bn * Gn * NTAP;        // B*G*25

    // 1) 1x1 conv as fp32 WMMA GEMM
    conv_gemm_wmma<<<dim3(COUT / 16, HW / 64, Bn), 32, 0, stream>>>(
        x, conv_w, conv_b, attn);

    // 2) GroupNorm window statistics (mean / rstd per (b,g,kh,kw))
    gn_stats<<<dim3(Gn, Bn), 256, 0, stream>>>(attn, meanA, rstdA);

    // 3) Fused GN-apply + 5x5 local attention
    local_attn<<<dim3(Hh, NHn, Bn), 256, 0, stream>>>(
        attn, meanA, rstdA, gn_w, gn_b, out);
}